// Encoder_57836029608226
// MI455X (gfx1250) — compile-verified
//
#include <hip/hip_runtime.h>
#include <hip/hip_bf16.h>

// ---------------------------------------------------------------------------
// CDNA5 (gfx1250) wave32 WMMA helpers: V_WMMA_F32_16X16X4_F32
//   A: 16x4 f32  -> 2 VGPR/lane (lanes 0-15: M=lane, K=k0..k0+1; lanes 16-31: K=k0+2..k0+3)
//   B: 4x16 f32  -> 2 VGPR/lane (lanes 0-15: N=lane, K=k0..;     lanes 16-31: K=k0+2..)
//   C/D: 16x16   -> 8 VGPR/lane, element (M = v + 8*(lane>=16), N = lane&15)
// ---------------------------------------------------------------------------
typedef float v2f __attribute__((ext_vector_type(2)));
typedef float v8f __attribute__((ext_vector_type(8)));

__device__ __forceinline__ v8f wmma4(v2f a, v2f b, v8f c) {
  return __builtin_amdgcn_wmma_f32_16x16x4_f32(false, a, false, b, (short)0, c,
                                               false, false);
}

#define NB    32
#define RR    500
#define DM    1024
#define DKV   16
#define DFF   256
#define BR    (NB * RR)   // 16000 rows, 1000 16-row tiles

// ---------------------------------------------------------------------------
// Kernel 1: link-feature MLPs: lf = sig(MLP1(v1)) + sig(MLP2(v2))   [B, DM]
// ---------------------------------------------------------------------------
__global__ void k_linkfeat(const float* __restrict__ v1, const float* __restrict__ v2,
                           const float* W1a, const float* W2a, const float* W3a,
                           const float* b1a, const float* b2a, const float* b3a,
                           const float* W1b, const float* W2b, const float* W3b,
                           const float* b1b, const float* b2b, const float* b3b,
                           float* __restrict__ lf) {
  __shared__ float vin[DM];
  __shared__ float h1[100];
  __shared__ float h2[100];
  int b = blockIdx.x, tid = threadIdx.x;
  float out[4] = {0.f, 0.f, 0.f, 0.f};
  for (int m = 0; m < 2; ++m) {
    const float* vsrc = m ? v2 : v1;
    const float* W1 = m ? W1b : W1a;  const float* W2 = m ? W2b : W2a;
    const float* W3 = m ? W3b : W3a;  const float* b1 = m ? b1b : b1a;
    const float* b2 = m ? b2b : b2a;  const float* b3 = m ? b3b : b3a;
    for (int j = tid; j < DM; j += 256) vin[j] = vsrc[b * DM + j];
    __syncthreads();
    if (tid < 100) {
      float a = b1[tid];
      for (int k = 0; k < DM; ++k) a += W1[tid * DM + k] * vin[k];
      h1[tid] = fmaxf(a, 0.f);
    }
    __syncthreads();
    if (tid < 100) {
      float a = b2[tid];
      for (int k = 0; k < 100; ++k) a += W2[tid * 100 + k] * h1[k];
      h2[tid] = fmaxf(a, 0.f);
    }
    __syncthreads();
    for (int q = 0; q < 4; ++q) {
      int j = tid + 256 * q;
      float a = b3[j];
      for (int k = 0; k < 100; ++k) a += W3[j * 100 + k] * h2[k];
      out[q] += 1.f / (1.f + expf(-a));
    }
    __syncthreads();
  }
  for (int q = 0; q < 4; ++q) lf[b * DM + tid + 256 * q] = out[q];
}

// ---------------------------------------------------------------------------
// Kernel 2: fold dimc:  w_eff[j] = sum_n Wp[n]*Wh[n,j];  b_eff = Wp.bh + bp
// ---------------------------------------------------------------------------
__global__ void k_weff(const float* __restrict__ Wh, const float* __restrict__ Wp,
                       const float* __restrict__ bh, const float* __restrict__ bp,
                       float* __restrict__ weff, float* __restrict__ beff) {
  int j = blockIdx.x * 256 + threadIdx.x;
  if (j < DM) {
    float a = 0.f;
    for (int n = 0; n < 512; ++n) a += Wp[n] * Wh[n * DM + j];
    weff[j] = a;
  }
  if (blockIdx.x == 0 && threadIdx.x == 0) {
    float a = bp[0];
    for (int n = 0; n < 512; ++n) a += Wp[n] * bh[n];
    beff[0] = a;
  }
}

// ---------------------------------------------------------------------------
// Kernel 3: x = enc + lf + PE  (sinusoidal PE on the fly); keymask from s==0
// ---------------------------------------------------------------------------
__global__ void k_buildx(const float* __restrict__ enc, const float* __restrict__ lf,
                         const float* __restrict__ weff, const float* __restrict__ beff,
                         float* __restrict__ xpe, float* __restrict__ keymask) {
  __shared__ float red[256];
  long row = blockIdx.x;
  int b = (int)(row / RR), r = (int)(row % RR), tid = threadIdx.x;
  const float cdiv = -9.210340371976184f / 1024.f;  // -ln(10000)/DM
  float part = 0.f;
  for (int q = 0; q < 4; ++q) {
    int j = tid + 256 * q;
    float x0 = enc[row * DM + j] + lf[b * DM + j];
    part += x0 * weff[j];
    float ang = (float)r * expf((float)(j & ~1) * cdiv);
    float pe = (j & 1) ? cosf(ang) : sinf(ang);
    xpe[row * DM + j] = x0 + pe;
  }
  red[tid] = part;
  __syncthreads();
  for (int s = 128; s > 0; s >>= 1) {
    if (tid < s) red[tid] += red[tid + s];
    __syncthreads();
  }
  if (tid == 0) {
    float sv = red[0] + beff[0];
    keymask[row] = (sv == 0.f) ? 1.f : 0.f;
  }
}

// ---------------------------------------------------------------------------
// Kernel 4: fused QKV projection [BR,1024]x[1024,16] via WMMA f32 16x16x4.
// One 16-row M-tile per wave. Software-pipelined: 32 loads, then 24 WMMAs
// per chunk so loadcnt waits overlap with the matrix pipe.
// ---------------------------------------------------------------------------
__global__ void k_qkv(const float* __restrict__ xpe,
                      const float* __restrict__ Wq, const float* __restrict__ Wk,
                      const float* __restrict__ Wv,
                      float* __restrict__ Q, float* __restrict__ K, float* __restrict__ V) {
  int tid = threadIdx.x;
  int wave = tid >> 5, lane = tid & 31;
  int half = lane >> 4, l16 = lane & 15;
  long m0 = ((long)blockIdx.x * 8 + wave) * 16;
  int koff = 2 * half;
  v8f cq = {}, ck = {}, cv = {};
  const float* arow = xpe + (m0 + l16) * DM + koff;
  const float* bqr = Wq + l16 * DM + koff;
  const float* bkr = Wk + l16 * DM + koff;
  const float* bvr = Wv + l16 * DM + koff;
  for (int k0 = 0; k0 < DM; k0 += 32) {
    v2f a[8], bq[8], bk[8], bv[8];
#pragma unroll
    for (int i = 0; i < 8; ++i) {
      int o = k0 + 4 * i;
      a[i]  = *(const v2f*)(arow + o);
      bq[i] = *(const v2f*)(bqr + o);
      bk[i] = *(const v2f*)(bkr + o);
      bv[i] = *(const v2f*)(bvr + o);
    }
#pragma unroll
    for (int i = 0; i < 8; ++i) {
      cq = wmma4(a[i], bq[i], cq);
      ck = wmma4(a[i], bk[i], ck);
      cv = wmma4(a[i], bv[i], cv);
    }
  }
  for (int v = 0; v < 8; ++v) {
    long m = m0 + v + 8 * half;
    Q[m * DKV + l16] = cq[v];
    K[m * DKV + l16] = ck[v];
    V[m * DKV + l16] = cv[v];
  }
}

// ---------------------------------------------------------------------------
// Kernel 5: attention for one (batch, 16-q-row tile). 4 waves / block.
// scores (WMMA) -> +conv(sim) -> mask -> softmax (LDS) -> attn out -> P@V (WMMA)
// ---------------------------------------------------------------------------
__global__ void __launch_bounds__(128)
k_attn(const float* __restrict__ Q, const float* __restrict__ K,
       const float* __restrict__ V, const float* __restrict__ sim,
       const float* __restrict__ keymask, const float* __restrict__ convw,
       const float* __restrict__ convb, float* __restrict__ attn_out,
       float* __restrict__ ctx) {
  __shared__ float sS[16][520];      // 16 q-rows x 512 k (padded)
  __shared__ float red[16][8];
  __shared__ float rowmax[16], rowsum[16];
  __shared__ float pc[4][16][16];
  int qt = blockIdx.x, b = blockIdx.y;
  int tid = threadIdx.x, wave = tid >> 5, lane = tid & 31;
  int half = lane >> 4, l16 = lane & 15;
  int q0 = qt * 16;
  float cw = convw[0], cb = convb[0];
  const float scale = 0.25f;  // 1/sqrt(16)

  int qr = q0 + l16;
  int qrc = qr < RR ? qr : (RR - 1);
  v2f afrag[4];
#pragma unroll
  for (int kk = 0; kk < 4; ++kk)
    afrag[kk] = *(const v2f*)(Q + ((long)b * RR + qrc) * DKV + kk * 4 + 2 * half);

  for (int kt = wave; kt < 32; kt += 4) {
    int kbase = kt * 16;
    int krow = kbase + l16;
    int krc = krow < RR ? krow : (RR - 1);
    v2f bf[4];
#pragma unroll
    for (int kk = 0; kk < 4; ++kk)
      bf[kk] = *(const v2f*)(K + ((long)b * RR + krc) * DKV + kk * 4 + 2 * half);
    v8f c = {};
#pragma unroll
    for (int kk = 0; kk < 4; ++kk) c = wmma4(afrag[kk], bf[kk], c);
    for (int v = 0; v < 8; ++v) {
      int m = v + 8 * half;
      int q = q0 + m;
      int kcol = kbase + l16;
      float sc;
      if (q >= RR) sc = 0.f;                    // dead rows: keep finite
      else if (kcol >= RR) sc = -INFINITY;      // pad cols: exp -> 0
      else {
        sc = c[v] * scale + cw * sim[((long)b * RR + q) * RR + kcol] + cb;
        if (keymask[b * RR + kcol] != 0.f) sc = -1e9f;
      }
      sS[m][kbase + l16] = sc;
    }
  }
  __syncthreads();

  // softmax: 8 threads per row
  int r = tid >> 3, sub = tid & 7;
  float mx = -INFINITY;
  for (int j = sub; j < 512; j += 8) mx = fmaxf(mx, sS[r][j]);
  red[r][sub] = mx;
  __syncthreads();
  if (sub == 0) {
    float m = red[r][0];
    for (int i = 1; i < 8; ++i) m = fmaxf(m, red[r][i]);
    rowmax[r] = m;
  }
  __syncthreads();
  float rm = rowmax[r], sum = 0.f;
  for (int j = sub; j < 512; j += 8) {
    float e = expf(sS[r][j] - rm);
    sS[r][j] = e;
    sum += e;
  }
  red[r][sub] = sum;
  __syncthreads();
  if (sub == 0) {
    float s2 = 0.f;
    for (int i = 0; i < 8; ++i) s2 += red[r][i];
    rowsum[r] = s2;
  }
  __syncthreads();
  float inv = 1.f / rowsum[r];
  for (int j = sub; j < 512; j += 8) sS[r][j] *= inv;
  __syncthreads();

  // write attention probabilities
  for (int m = 0; m < 16; ++m) {
    int q = q0 + m;
    if (q < RR)
      for (int k = tid; k < RR; k += 128)
        attn_out[((long)b * RR + q) * RR + k] = sS[m][k];
  }

  // ctx = P @ V via WMMA; each wave covers 8 k-tiles, partials reduced in LDS
  v8f c = {};
  for (int kt = wave; kt < 32; kt += 4) {
    int kbase = kt * 16;
    v2f a[4], bf[4];
#pragma unroll
    for (int kk = 0; kk < 4; ++kk) {
      int k0 = kbase + kk * 4;
      int kg = k0 + 2 * half;
      int kg0 = kg < RR ? kg : (RR - 1);
      int kg1 = (kg + 1) < RR ? (kg + 1) : (RR - 1);
      bf[kk].x = V[((long)b * RR + kg0) * DKV + l16];
      bf[kk].y = V[((long)b * RR + kg1) * DKV + l16];
      a[kk] = *(const v2f*)&sS[l16][k0 + 2 * half];
    }
#pragma unroll
    for (int kk = 0; kk < 4; ++kk) c = wmma4(a[kk], bf[kk], c);
  }
  for (int v = 0; v < 8; ++v) pc[wave][v + 8 * half][l16] = c[v];
  __syncthreads();
  for (int idx = tid; idx < 256; idx += 128) {
    int m = idx >> 4, n = idx & 15;
    int q = q0 + m;
    if (q < RR)
      ctx[((long)b * RR + q) * DKV + n] =
          pc[0][m][n] + pc[1][m][n] + pc[2][m][n] + pc[3][m][n];
  }
}

// ---------------------------------------------------------------------------
// Kernel 6: x1 = LN(ctx @ Wo^T + x)   (block per row, VALU)
// ---------------------------------------------------------------------------
__global__ void k_attnout(const float* __restrict__ ctx, const float* __restrict__ Wo,
                          const float* __restrict__ xpe, float* __restrict__ xout) {
  __shared__ float cs[16];
  __shared__ float red[256];
  long row = blockIdx.x;
  int tid = threadIdx.x;
  if (tid < 16) cs[tid] = ctx[row * DKV + tid];
  __syncthreads();
  float t[4], psum = 0.f;
  for (int q = 0; q < 4; ++q) {
    int j = tid + 256 * q;
    float a = xpe[row * DM + j];
    for (int d = 0; d < 16; ++d) a += cs[d] * Wo[j * 16 + d];
    t[q] = a;
    psum += a;
  }
  red[tid] = psum;
  __syncthreads();
  for (int s = 128; s > 0; s >>= 1) {
    if (tid < s) red[tid] += red[tid + s];
    __syncthreads();
  }
  float mean = red[0] * (1.f / 1024.f);
  __syncthreads();
  float pv = 0.f;
  for (int q = 0; q < 4; ++q) {
    float d = t[q] - mean;
    pv += d * d;
  }
  red[tid] = pv;
  __syncthreads();
  for (int s = 128; s > 0; s >>= 1) {
    if (tid < s) red[tid] += red[tid + s];
    __syncthreads();
  }
  float rstd = rsqrtf(red[0] * (1.f / 1024.f) + 1e-5f);
  for (int q = 0; q < 4; ++q)
    xout[row * DM + tid + 256 * q] = (t[q] - mean) * rstd;
}

// ---------------------------------------------------------------------------
// Kernel 7: h = relu(x1 @ Wf1^T)   [BR,1024]x[1024,256]; A staged in LDS,
// pipelined chunks of 8 K-steps (DS + VMEM latencies overlap the WMMAs).
// ---------------------------------------------------------------------------
__global__ void k_ffn1(const float* __restrict__ x, const float* __restrict__ Wf1,
                       float* __restrict__ h) {
  __shared__ float xs[16][DM];  // 64 KB of the 320 KB WGP LDS
  int tid = threadIdx.x;
  long m0 = (long)blockIdx.x * 16;
  for (int idx = tid; idx < 4096; idx += 256) {
    int m = idx >> 8, c4 = idx & 255;
    *(float4*)&xs[m][c4 * 4] = *(const float4*)(x + (m0 + m) * DM + c4 * 4);
  }
  __syncthreads();
  int wave = tid >> 5, lane = tid & 31, half = lane >> 4, l16 = lane & 15;
  for (int j = 0; j < 2; ++j) {
    int n0 = (wave * 2 + j) * 16;
    v8f c = {};
    const float* brow = Wf1 + (n0 + l16) * DM + 2 * half;
    for (int k0 = 0; k0 < DM; k0 += 32) {
      v2f a[8], bf[8];
#pragma unroll
      for (int i = 0; i < 8; ++i) {
        int o = k0 + 4 * i;
        bf[i] = *(const v2f*)(brow + o);
        a[i]  = *(const v2f*)&xs[l16][o + 2 * half];
      }
#pragma unroll
      for (int i = 0; i < 8; ++i) c = wmma4(a[i], bf[i], c);
    }
    for (int v = 0; v < 8; ++v)
      h[(m0 + v + 8 * half) * DFF + n0 + l16] = fmaxf(c[v], 0.f);
  }
}

// ---------------------------------------------------------------------------
// Kernel 8: x = LN(h @ Wf2^T + x1)  in-place; row stats via ds_add_f32 atomics
// ---------------------------------------------------------------------------
__global__ void k_ffn2(const float* __restrict__ h, const float* __restrict__ Wf2,
                       float* __restrict__ x) {
  __shared__ float hs[16][DFF];
  __shared__ float rsum[16];
  int tid = threadIdx.x;
  long m0 = (long)blockIdx.x * 16;
  for (int idx = tid; idx < 1024; idx += 256) {
    int m = idx >> 6, c4 = idx & 63;
    *(float4*)&hs[m][c4 * 4] = *(const float4*)(h + (m0 + m) * DFF + c4 * 4);
  }
  if (tid < 16) rsum[tid] = 0.f;
  __syncthreads();
  int wave = tid >> 5, lane = tid & 31, half = lane >> 4, l16 = lane & 15;
  v8f acc[8];
  for (int j = 0; j < 8; ++j) {
    int n0 = (wave * 8 + j) * 16;
    v8f c = {};
    const float* brow = Wf2 + (n0 + l16) * DFF + 2 * half;
    for (int k0 = 0; k0 < DFF; k0 += 32) {
      v2f a[8], bf[8];
#pragma unroll
      for (int i = 0; i < 8; ++i) {
        int o = k0 + 4 * i;
        bf[i] = *(const v2f*)(brow + o);
        a[i]  = *(const v2f*)&hs[l16][o + 2 * half];
      }
#pragma unroll
      for (int i = 0; i < 8; ++i) c = wmma4(a[i], bf[i], c);
    }
    for (int v = 0; v < 8; ++v)
      c[v] += x[(m0 + v + 8 * half) * DM + n0 + l16];  // residual
    acc[j] = c;
  }
  // mean
  for (int v = 0; v < 8; ++v) {
    float p = 0.f;
    for (int j = 0; j < 8; ++j) p += acc[j][v];
    atomicAdd(&rsum[v + 8 * half], p);  // ds_add_f32
  }
  __syncthreads();
  float mymean[8];
  for (int v = 0; v < 8; ++v) mymean[v] = rsum[v + 8 * half] * (1.f / 1024.f);
  __syncthreads();
  if (tid < 16) rsum[tid] = 0.f;
  __syncthreads();
  // variance
  for (int v = 0; v < 8; ++v) {
    float p = 0.f;
    for (int j = 0; j < 8; ++j) {
      float d = acc[j][v] - mymean[v];
      p += d * d;
    }
    atomicAdd(&rsum[v + 8 * half], p);
  }
  __syncthreads();
  float rstd[8];
  for (int v = 0; v < 8; ++v)
    rstd[v] = rsqrtf(rsum[v + 8 * half] * (1.f / 1024.f) + 1e-5f);
  for (int j = 0; j < 8; ++j) {
    int n0 = (wave * 8 + j) * 16;
    for (int v = 0; v < 8; ++v)
      x[(m0 + v + 8 * half) * DM + n0 + l16] = (acc[j][v] - mymean[v]) * rstd[v];
  }
}

// ---------------------------------------------------------------------------
extern "C" void kernel_launch(void* const* d_in, const int* in_sizes, int n_in,
                              void* d_out, int out_size, void* d_ws, size_t ws_size,
                              hipStream_t stream) {
  (void)in_sizes; (void)n_in; (void)out_size; (void)ws_size;
  // tree-flattened params (dict keys sorted: dimc, layers, le1, le2; 'W' < 'b')
  const float* enc   = (const float*)d_in[0];
  const float* sim   = (const float*)d_in[1];
  const float* v1    = (const float*)d_in[2];
  const float* v2    = (const float*)d_in[3];
  const float* Wh    = (const float*)d_in[4];   // dimc.Wh [512,1024]
  const float* Wp    = (const float*)d_in[5];   // dimc.Wp [1,512]
  const float* bh    = (const float*)d_in[6];
  const float* bp    = (const float*)d_in[7];
  const float* Wf1   = (const float*)d_in[8];   // [256,1024]
  const float* Wf2   = (const float*)d_in[9];   // [1024,256]
  const float* Wk    = (const float*)d_in[10];  // [16,1024]
  const float* Wo    = (const float*)d_in[11];  // [1024,16]
  const float* Wq    = (const float*)d_in[12];
  const float* Wv    = (const float*)d_in[13];
  const float* convb = (const float*)d_in[14];
  const float* convw = (const float*)d_in[15];
  const float* W1a = (const float*)d_in[16];
  const float* W2a = (const float*)d_in[17];
  const float* W3a = (const float*)d_in[18];
  const float* b1a = (const float*)d_in[19];
  const float* b2a = (const float*)d_in[20];
  const float* b3a = (const float*)d_in[21];
  const float* W1b = (const float*)d_in[22];
  const float* W2b = (const float*)d_in[23];
  const float* W3b = (const float*)d_in[24];
  const float* b1b = (const float*)d_in[25];
  const float* b2b = (const float*)d_in[26];
  const float* b3b = (const float*)d_in[27];

  float* ws = (float*)d_ws;
  float* xpe     = ws;                       // 16,384,000
  float* lf      = xpe + 16384000;           // 32,768
  float* weff    = lf + 32768;               // 1,024
  float* beff    = weff + 1024;              // 8
  float* keymask = beff + 8;                 // 16,008
  float* Qm      = keymask + 16008;          // 256,000
  float* Km      = Qm + 256000;              // 256,000
  float* Vm      = Km + 256000;              // 256,000
  float* ctx     = Vm + 256000;              // 256,000
  float* hff     = ctx + 256000;             // 4,096,000  (~87 MB total)

  float* xout    = (float*)d_out;            // 16,384,000
  float* attn    = xout + 16384000;          // 8,000,000
  float* simout  = attn + 8000000;           // 8,000,000

  k_linkfeat<<<NB, 256, 0, stream>>>(v1, v2, W1a, W2a, W3a, b1a, b2a, b3a,
                                     W1b, W2b, W3b, b1b, b2b, b3b, lf);
  k_weff<<<4, 256, 0, stream>>>(Wh, Wp, bh, bp, weff, beff);
  k_buildx<<<BR, 256, 0, stream>>>(enc, lf, weff, beff, xpe, keymask);
  k_qkv<<<BR / 128, 256, 0, stream>>>(xpe, Wq, Wk, Wv, Qm, Km, Vm);
  k_attn<<<dim3(32, NB), 128, 0, stream>>>(Qm, Km, Vm, sim, keymask, convw,
                                           convb, attn, ctx);
  k_attnout<<<BR, 256, 0, stream>>>(ctx, Wo, xpe, xout);
  k_ffn1<<<BR / 16, 256, 0, stream>>>(xout, Wf1, hff);
  k_ffn2<<<BR / 16, 256, 0, stream>>>(hff, Wf2, xout);
  hipMemcpyAsync(simout, sim, (size_t)8000000 * sizeof(float),
                 hipMemcpyDeviceToDevice, stream);
}